// BFMFaceLoss_50835232915523
// MI455X (gfx1250) — compile-verified
//
#include <hip/hip_runtime.h>
#include <math.h>

// ---------------------------------------------------------------------------
// BFM face shading: face normals -> vertex normals -> SH lighting * albedo.
// Memory-bound (~190 MB @ 23.3 TB/s ~ 8us). The SH einsum (V x 9) @ (9 x 3)
// per batch is mapped onto V_WMMA_F32_16X16X4_F32 (three K=4 accumulations).
// ---------------------------------------------------------------------------

#define V_CNT 35709
#define F_CNT 70789
#define BS_CNT 64

typedef __attribute__((ext_vector_type(2))) float v2f;
typedef __attribute__((ext_vector_type(8))) float v8f;

#define PI_F 3.14159265358979323846f

// Kernel 1: face normals into workspace, laid out [BS][F+1][3], row F zeroed.
__global__ __launch_bounds__(256) void face_norm_kernel(
    const float* __restrict__ verts,   // [BS][V][3]
    const int*   __restrict__ tri,     // [3*F]
    float*       __restrict__ fnorm)   // [BS][F+1][3]
{
    const int f = blockIdx.x * blockDim.x + threadIdx.x;
    const int b = blockIdx.y;
    if (f > F_CNT) return;

    float* o = fnorm + ((size_t)b * (F_CNT + 1) + f) * 3;
    if (f == F_CNT) {                      // appended zero row (pid == F)
        o[0] = 0.0f; o[1] = 0.0f; o[2] = 0.0f;
        return;
    }
    // face_id = flip(tri.reshape(-1,3) - 1): [tri[3f+2]-1, tri[3f+1]-1, tri[3f]-1]
    const int i0 = tri[3 * f + 2] - 1;
    const int i1 = tri[3 * f + 1] - 1;
    const int i2 = tri[3 * f + 0] - 1;

    const float* p0 = verts + ((size_t)b * V_CNT + i0) * 3;
    const float* p1 = verts + ((size_t)b * V_CNT + i1) * 3;
    const float* p2 = verts + ((size_t)b * V_CNT + i2) * 3;

    const float ax = p0[0] - p1[0], ay = p0[1] - p1[1], az = p0[2] - p1[2];
    const float bx = p1[0] - p2[0], by = p1[1] - p2[1], bz = p1[2] - p2[2];

    o[0] = ay * bz - az * by;
    o[1] = az * bx - ax * bz;
    o[2] = ax * by - ay * bx;
}

// Kernel 2: per 16-vertex tile per wave: gather 8 face normals, normalize,
// build SH basis Y (K=9, zero-padded to 12), then
//   D(16x16) += A(16x4) * B(4x16)   three times (K chunks 0..3, 4..7, 8..11)
// with B = gamma matrix for this batch in columns N=0..2, zeros elsewhere.
__global__ __launch_bounds__(256) void shade_kernel(
    const float* __restrict__ fnorm,     // [BS][F+1][3]
    const int*   __restrict__ point_buf, // [V][8]
    const float* __restrict__ albedo,    // [BS][V][3]
    const float* __restrict__ gamma,     // [BS][27]
    float*       __restrict__ out)       // [BS][V][3]
{
    const int lane = threadIdx.x & 31;
    const int wave = threadIdx.x >> 5;
    const int b = blockIdx.y;
    const int vbase = (blockIdx.x * 8 + wave) * 16;   // 16 vertices per wave
    const bool hi = lane >= 16;

    // ---- vertex normal for vertex vbase+(lane&15); both half-waves compute it
    //      (ISA A layout: lanes 0-15 carry K0/K1, lanes 16-31 carry K2/K3 of
    //       the SAME row M = lane&15) ----
    const int v = vbase + (lane & 15);
    const int vv = v < V_CNT ? v : V_CNT - 1;          // clamp loads; rows >= V masked at store
    const int4* pb4 = (const int4*)(point_buf + (size_t)vv * 8);
    const int4 pa = pb4[0];
    const int4 pb = pb4[1];
    const float* fb = fnorm + (size_t)b * (F_CNT + 1) * 3;

    float sx = 0.0f, sy = 0.0f, sz = 0.0f;
    auto gath = [&](int pointv) {
        const float* fn = fb + (size_t)(pointv - 1) * 3;   // pid in [0,F]; row F is zeros
        sx += fn[0]; sy += fn[1]; sz += fn[2];
    };
    gath(pa.x); gath(pa.y); gath(pa.z); gath(pa.w);
    gath(pb.x); gath(pb.y); gath(pb.z); gath(pb.w);

    const float len = sqrtf(sx * sx + sy * sy + sz * sz);
    const float inv = 1.0f / fmaxf(len, 1e-12f);
    const float nx = sx * inv, ny = sy * inv, nz = sz * inv;

    // ---- SH basis coefficients (constant-folded) ----
    const float c0 = 1.0f / sqrtf(4.0f * PI_F);
    const float c1 = sqrtf(3.0f) / sqrtf(4.0f * PI_F);
    const float c2 = 3.0f * sqrtf(5.0f) / sqrtf(12.0f * PI_F);
    const float a0c0 = PI_F * c0;
    const float a1c1 = (2.0f * PI_F / sqrtf(3.0f)) * c1;
    const float a2c2 = (2.0f * PI_F / sqrtf(8.0f)) * c2;

    const float y0 = a0c0;
    const float y1 = -a1c1 * ny;
    const float y2 =  a1c1 * nz;
    const float y3 = -a1c1 * nx;
    const float y4 =  a2c2 * nx * ny;
    const float y5 = -a2c2 * ny * nz;
    const float y6 =  a2c2 * (0.5f / sqrtf(3.0f)) * (3.0f * nz * nz - 1.0f);
    const float y7 = -a2c2 * nx * nz;
    const float y8 =  a2c2 * 0.5f * (nx * nx - ny * ny);

    // ---- B operand: column n = lane&15 holds gamma(b, n, k) for n < 3 ----
    const int n = lane & 15;
    const int nc = n < 3 ? n : 2;                       // clamp loads
    const float sel = n < 3 ? 1.0f : 0.0f;              // mask columns 3..15 to zero
    const float* gp = gamma + (size_t)b * 27 + nc * 9;
    const float g0 = (gp[0] + 0.8f) * sel;              // init_light[0] = 0.8
    const float g1 = gp[1] * sel;
    const float g2 = gp[2] * sel;
    const float g3 = gp[3] * sel;
    const float g4 = gp[4] * sel;
    const float g5 = gp[5] * sel;
    const float g6 = gp[6] * sel;
    const float g7 = gp[7] * sel;
    const float g8 = gp[8] * sel;

    // ---- three chained V_WMMA_F32_16X16X4_F32 accumulations ----
    // A (16x4 f32): lanes 0-15 -> {K0,K1}, lanes 16-31 -> {K2,K3} of row M=lane&15.
    // B (4x16 f32): lanes 0-15 -> {K0,K1}, lanes 16-31 -> {K2,K3} of col N=lane&15.
    v2f A0, B0, A1, B1, A2, B2;
    A0.x = hi ? y2 : y0;   A0.y = hi ? y3 : y1;
    B0.x = hi ? g2 : g0;   B0.y = hi ? g3 : g1;
    A1.x = hi ? y6 : y4;   A1.y = hi ? y7 : y5;
    B1.x = hi ? g6 : g4;   B1.y = hi ? g7 : g5;
    A2.x = hi ? 0.0f : y8; A2.y = 0.0f;                 // K=8 only; K9..11 are zero pad
    B2.x = hi ? 0.0f : g8; B2.y = 0.0f;

    v8f acc = {};
    acc = __builtin_amdgcn_wmma_f32_16x16x4_f32(false, A0, false, B0, (short)0, acc, false, false);
    acc = __builtin_amdgcn_wmma_f32_16x16x4_f32(false, A1, false, B1, (short)0, acc, false, false);
    acc = __builtin_amdgcn_wmma_f32_16x16x4_f32(false, A2, false, B2, (short)0, acc, false, false);

    // ---- D layout: VGPR r holds row M = r + (hi ? 8 : 0), column N = lane&15 ----
    if (n < 3) {
        const int rbase = vbase + (hi ? 8 : 0);
        const size_t base = ((size_t)b * V_CNT + rbase) * 3 + n;
        const float* ap = albedo + base;                // row stride = 3 floats
        float*       op = out + base;                   // constant offsets fold into insts
        if (rbase + 8 <= V_CNT) {                       // fast path: full tile, no guards
#pragma unroll
            for (int r = 0; r < 8; ++r)
                op[3 * r] = acc[r] * ap[3 * r];
        } else {                                        // last partial tile only
#pragma unroll
            for (int r = 0; r < 8; ++r)
                if (rbase + r < V_CNT)
                    op[3 * r] = acc[r] * ap[3 * r];
        }
    }
}

extern "C" void kernel_launch(void* const* d_in, const int* in_sizes, int n_in,
                              void* d_out, int out_size, void* d_ws, size_t ws_size,
                              hipStream_t stream) {
    (void)in_sizes; (void)n_in; (void)out_size; (void)ws_size;

    const float* vertices  = (const float*)d_in[0];  // [BS][V][3]
    const float* albedo    = (const float*)d_in[1];  // [BS][V][3]
    const float* gamma     = (const float*)d_in[2];  // [BS][27]
    const int*   tri       = (const int*)d_in[3];    // [3*F]
    const int*   point_buf = (const int*)d_in[4];    // [V][8]
    float*       out       = (float*)d_out;          // [BS][V][3]
    float*       fnorm     = (float*)d_ws;           // [BS][F+1][3] scratch (~52 MB)

    {
        dim3 blk(256);
        dim3 grd((F_CNT + 1 + 255) / 256, BS_CNT);
        face_norm_kernel<<<grd, blk, 0, stream>>>(vertices, tri, fnorm);
    }
    {
        dim3 blk(256);                                // 8 waves * 16 vertices = 128 / block
        dim3 grd((V_CNT + 127) / 128, BS_CNT);
        shade_kernel<<<grd, blk, 0, stream>>>(fnorm, point_buf, albedo, gamma, out);
    }
}